// MoELayer_80169859548016
// MI455X (gfx1250) — compile-verified
//
#include <hip/hip_runtime.h>
#include <stdint.h>

// Problem constants (B=2, S=2048 -> N=4096 tokens)
#define NTOK 4096
#define DDIM 1024
#define HDIM 2048
#define NEXP 8

typedef float v2f __attribute__((ext_vector_type(2)));
typedef float v8f __attribute__((ext_vector_type(8)));

// ---------------------------------------------------------------------------
// 1) Gating: logits = x @ gate_w + gate_b ; top-2 ; softmax(top2) ; aux term
// ---------------------------------------------------------------------------
__global__ void gate_kernel(const float* __restrict__ x,
                            const float* __restrict__ gw,
                            const float* __restrict__ gb,
                            int* __restrict__ idx0, int* __restrict__ idx1,
                            float* __restrict__ s0, float* __restrict__ s1,
                            float* __restrict__ auxtok)
{
    __shared__ float part[128][NEXP];
    const int n   = blockIdx.x;      // token
    const int tid = threadIdx.x;     // 0..127

    float p[NEXP];
#pragma unroll
    for (int e = 0; e < NEXP; ++e) p[e] = 0.f;

    const float* xr = x + (size_t)n * DDIM;
    for (int d = tid; d < DDIM; d += 128) {
        const float xv = xr[d];
        const float* g = gw + (size_t)d * NEXP;
#pragma unroll
        for (int e = 0; e < NEXP; ++e) p[e] += xv * g[e];
    }
#pragma unroll
    for (int e = 0; e < NEXP; ++e) part[tid][e] = p[e];
    __syncthreads();
    for (int s = 64; s > 0; s >>= 1) {
        if (tid < s) {
#pragma unroll
            for (int e = 0; e < NEXP; ++e) part[tid][e] += part[tid + s][e];
        }
        __syncthreads();
    }
    if (tid == 0) {
        float lg[NEXP];
#pragma unroll
        for (int e = 0; e < NEXP; ++e) lg[e] = part[0][e] + gb[e];
        // top-2, ties -> lowest index (matches jax.lax.top_k)
        int i0 = 0;
#pragma unroll
        for (int e = 1; e < NEXP; ++e) if (lg[e] > lg[i0]) i0 = e;
        int i1 = (i0 == 0) ? 1 : 0;
#pragma unroll
        for (int e = 0; e < NEXP; ++e)
            if (e != i0 && lg[e] > lg[i1]) i1 = e;

        const float m  = fmaxf(lg[i0], lg[i1]);
        const float e0 = expf(lg[i0] - m);
        const float e1 = expf(lg[i1] - m);
        const float iv = 1.f / (e0 + e1);
        idx0[n] = i0; idx1[n] = i1;
        s0[n] = e0 * iv; s1[n] = e1 * iv;

        // aux: sum_e (log(1/E) - logp_e) = E*log(1/E) - (sum(lg) - E*lse)
        float m8 = lg[0];
#pragma unroll
        for (int e = 1; e < NEXP; ++e) m8 = fmaxf(m8, lg[e]);
        float se = 0.f, sl = 0.f;
#pragma unroll
        for (int e = 0; e < NEXP; ++e) { se += expf(lg[e] - m8); sl += lg[e]; }
        const float lse = m8 + logf(se);
        auxtok[n] = (float)NEXP * logf(1.0f / (float)NEXP) - (sl - (float)NEXP * lse);
    }
}

// ---------------------------------------------------------------------------
// 2) Deterministic routing: per-expert compaction via block prefix scan
// ---------------------------------------------------------------------------
__global__ void route_kernel(const int* __restrict__ idx0, const int* __restrict__ idx1,
                             const float* __restrict__ s0, const float* __restrict__ s1,
                             int* __restrict__ counts, int* __restrict__ list,
                             float* __restrict__ lsc)
{
    const int e = blockIdx.x;        // expert
    __shared__ int scan[256];
    __shared__ int basec;
    const int tid = threadIdx.x;
    if (tid == 0) basec = 0;
    __syncthreads();

    for (int base = 0; base < NTOK; base += 256) {
        const int n = base + tid;
        int k = (idx0[n] == e) ? 0 : ((idx1[n] == e) ? 1 : -1);
        const int f = (k >= 0) ? 1 : 0;
        scan[tid] = f;
        __syncthreads();
        for (int off = 1; off < 256; off <<= 1) {
            int v = (tid >= off) ? scan[tid - off] : 0;
            __syncthreads();
            scan[tid] += v;
            __syncthreads();
        }
        if (f) {
            const int pos = basec + scan[tid] - 1;
            list[e * NTOK + pos] = (n << 1) | k;
            lsc [e * NTOK + pos] = k ? s1[n] : s0[n];
        }
        __syncthreads();
        if (tid == 0) basec += scan[255];
        __syncthreads();
    }
    if (tid == 0) counts[e] = basec;
}

// ---------------------------------------------------------------------------
// 3) Fused expert FFN: 1 WG = (expert, 16-token tile), 128 threads = 4 waves.
//    O[16,1024] = score * ( gelu(X[16,1024] @ W1[e]) @ W2[e] )
//    All global operands flow through the async-DMA -> LDS path:
//      * X rows: gathered global_load_async_to_lds_b128
//      * W1/W2 : per-wave double-buffered 8KB strips (ASYNCcnt ping-pong),
//                so WMMA B-fragments are LDS reads, never dependent VMEM.
// ---------------------------------------------------------------------------
__global__ void __launch_bounds__(128)
expert_kernel(const float* __restrict__ x,
              const float* __restrict__ w1,
              const float* __restrict__ w2,
              const int*   __restrict__ counts,
              const int*   __restrict__ list,
              const float* __restrict__ lsc,
              float* __restrict__ contrib)   // [2][NTOK][DDIM]
{
    const int e    = blockIdx.x >> 8;
    const int tile = blockIdx.x & 255;
    const int cnt  = counts[e];
    if (tile * 16 >= cnt) return;
    const int valid = (cnt - tile * 16 < 16) ? (cnt - tile * 16) : 16;

    __shared__ __align__(16) float Xs[16][DDIM];      // 64 KB token tile
    __shared__ __align__(16) float Hs[16][64];        // 4 KB H chunk
    __shared__ __align__(16) float Wb[4][2][2048];    // 64 KB: per-wave ping-pong strips
    __shared__ int   rowtok[16];
    __shared__ float rowsc[16];

    const int tid  = threadIdx.x;
    const int wave = tid >> 5;       // 0..3
    const int lane = tid & 31;

    if (tid < 16) {
        const int r = tid;
        const int i = tile * 16 + r;
        rowtok[r] = (r < valid) ? list[e * NTOK + i] : 0;  // dummy rows -> token 0 (never stored)
        rowsc[r]  = (r < valid) ? lsc[e * NTOK + i] : 0.f;
    }
    __syncthreads();

    // Gathered async copy of 16 token rows into LDS (per-lane addresses).
    const uint32_t xbase = (uint32_t)(uintptr_t)(&Xs[0][0]);
    for (int it = tid; it < 16 * (DDIM / 4); it += 128) {
        const int r  = it >> 8;       // row
        const int c4 = it & 255;      // which float4
        const int tok = rowtok[r] >> 1;
        const unsigned long long ga =
            (unsigned long long)(uintptr_t)(x + (size_t)tok * DDIM + c4 * 4);
        const uint32_t lo = xbase + (uint32_t)((r * DDIM + c4 * 4) * 4);
        asm volatile("global_load_async_to_lds_b128 %0, %1, off"
                     :: "v"(lo), "v"(ga) : "memory");
    }
    asm volatile("s_wait_asynccnt 0x0" ::: "memory");
    __syncthreads();

    const float* w1p = w1 + (size_t)e * DDIM * HDIM;
    const float* w2p = w2 + (size_t)e * HDIM * DDIM;

    // WMMA fragment indices (16x16x4 f32):
    //  A (16x4):  lanes 0-15: M=lane, K=koff..koff+1, koff=0 ; lanes 16-31: koff=2
    //  B (4x16):  lanes 0-15: N=lane ; same koff split
    //  C/D: acc[r] -> row (r + 8*(lane>=16)), col = lane&15
    const int row   = lane & 15;
    const int col   = lane & 15;
    const int koff  = (lane >> 4) << 1;          // 0 or 2
    const int rbase = (lane >> 4) << 3;          // 0 or 8

    const uint32_t wb0 = (uint32_t)(uintptr_t)(&Wb[wave][0][0]);
    const uint32_t wb1 = (uint32_t)(uintptr_t)(&Wb[wave][1][0]);

    // Issue one 8KB W1 strip: rows krow0..krow0+127, cols (hc + wave*16)..+16.
    // 512 lanes-of-16B -> 16 async instructions per wave.
    auto issueW1 = [&](int hc, int krow0, uint32_t dst) {
        const float* src = w1p + (size_t)krow0 * HDIM + hc + wave * 16;
        for (int it = lane; it < 512; it += 32) {
            const int kl = it >> 2;          // 0..127
            const int c4 = it & 3;           // 4 x b128 per 16-float row
            const unsigned long long ga =
                (unsigned long long)(uintptr_t)(src + (size_t)kl * HDIM + c4 * 4);
            const uint32_t lo = dst + (uint32_t)((kl * 16 + c4 * 4) * 4);
            asm volatile("global_load_async_to_lds_b128 %0, %1, off"
                         :: "v"(lo), "v"(ga) : "memory");
        }
    };
    // Issue one 8KB W2 strip: rows hc..hc+63, cols (wave*256 + g*32)..+32.
    auto issueW2 = [&](int hc, int g, uint32_t dst) {
        const float* src = w2p + (size_t)hc * DDIM + wave * 256 + g * 32;
        for (int it = lane; it < 512; it += 32) {
            const int kl = it >> 3;          // 0..63
            const int c4 = it & 7;           // 8 x b128 per 32-float row
            const unsigned long long ga =
                (unsigned long long)(uintptr_t)(src + (size_t)kl * DDIM + c4 * 4);
            const uint32_t lo = dst + (uint32_t)((kl * 32 + c4 * 4) * 4);
            asm volatile("global_load_async_to_lds_b128 %0, %1, off"
                         :: "v"(lo), "v"(ga) : "memory");
        }
    };

    v8f Oacc[16];
#pragma unroll
    for (int t = 0; t < 16; ++t) Oacc[t] = (v8f){0.f,0.f,0.f,0.f,0.f,0.f,0.f,0.f};

    for (int hc = 0; hc < HDIM; hc += 64) {
        // ---------- GEMM1: H[16, hc + wave*16 ..+16] = X @ W1 strip ----------
        issueW1(hc, 0, wb0);
        v8f acc = (v8f){0.f,0.f,0.f,0.f,0.f,0.f,0.f,0.f};
        for (int kb = 0; kb < 8; ++kb) {
            if (kb < 7) {
                issueW1(hc, (kb + 1) * 128, ((kb + 1) & 1) ? wb1 : wb0);
            } else {
                issueW2(hc, 0, wb0);        // prefetch first W2 strip (buf0 free)
            }
            asm volatile("s_wait_asynccnt 0x10" ::: "memory");  // strip kb resident
            float (*Wv)[16] = (float (*)[16])(&Wb[wave][kb & 1][0]);
            const int kbase = kb * 128;
            for (int j = 0; j < 128; j += 4) {
                v2f a = *(const v2f*)&Xs[row][kbase + j + koff];
                v2f b;
                b.x = Wv[j + koff][col];
                b.y = Wv[j + koff + 1][col];
                acc = __builtin_amdgcn_wmma_f32_16x16x4_f32(
                          false, a, false, b, (short)0, acc, false, false);
            }
        }
        // exact GELU, stage into LDS
#pragma unroll
        for (int r = 0; r < 8; ++r) {
            const float v = acc[r];
            Hs[r + rbase][wave * 16 + col] =
                0.5f * v * (1.0f + erff(v * 0.70710678118654752f));
        }
        __syncthreads();

        // ---------- GEMM2: O[16, wave*256..+256] += H chunk @ W2 strips ------
#pragma unroll
        for (int g = 0; g < 8; ++g) {
            if (g < 7) {
                issueW2(hc, g + 1, ((g + 1) & 1) ? wb1 : wb0);
                asm volatile("s_wait_asynccnt 0x10" ::: "memory"); // strip g ready
            } else {
                asm volatile("s_wait_asynccnt 0x0" ::: "memory");
            }
            float (*W2v)[32] = (float (*)[32])(&Wb[wave][g & 1][0]);
#pragma unroll
            for (int ti = 0; ti < 2; ++ti) {
                v8f o = Oacc[g * 2 + ti];
                for (int kk = 0; kk < 64; kk += 4) {
                    v2f a = *(const v2f*)&Hs[row][kk + koff];
                    v2f b;
                    b.x = W2v[kk + koff][ti * 16 + col];
                    b.y = W2v[kk + koff + 1][ti * 16 + col];
                    o = __builtin_amdgcn_wmma_f32_16x16x4_f32(
                            false, a, false, b, (short)0, o, false, false);
                }
                Oacc[g * 2 + ti] = o;
            }
        }
        __syncthreads();   // before Hs is overwritten next chunk
    }

    // Scale by gate score, scatter to per-slot plane (deterministic combine).
#pragma unroll
    for (int t = 0; t < 16; ++t) {
        const int n2 = wave * 256 + t * 16;
#pragma unroll
        for (int r = 0; r < 8; ++r) {
            const int rr = r + rbase;
            if (rr < valid) {
                const int packed = rowtok[rr];
                const int tok  = packed >> 1;
                const int slot = packed & 1;
                contrib[((size_t)slot * NTOK + tok) * DDIM + n2 + col] =
                    rowsc[rr] * Oacc[t][r];
            }
        }
    }
}

// ---------------------------------------------------------------------------
// 4) out = plane0 + plane1 (every token has exactly 2 assignments)
// ---------------------------------------------------------------------------
__global__ void combine_kernel(const float* __restrict__ contrib,
                               float* __restrict__ out)
{
    const size_t i = (size_t)blockIdx.x * blockDim.x + threadIdx.x; // float4 idx
    const float4* p0 = (const float4*)contrib;
    const float4* p1 = (const float4*)(contrib + (size_t)NTOK * DDIM);
    float4* o = (float4*)out;
    const float4 a = p0[i], b = p1[i];
    o[i] = make_float4(a.x + b.x, a.y + b.y, a.z + b.z, a.w + b.w);
}

// ---------------------------------------------------------------------------
// 5) aux loss: fixed-order tree reduction (deterministic)
// ---------------------------------------------------------------------------
__global__ void aux_kernel(const float* __restrict__ auxtok,
                           float* __restrict__ outaux)
{
    __shared__ float buf[256];
    const int tid = threadIdx.x;
    float s = 0.f;
    for (int i = tid; i < NTOK; i += 256) s += auxtok[i];
    buf[tid] = s;
    __syncthreads();
    for (int st = 128; st > 0; st >>= 1) {
        if (tid < st) buf[tid] += buf[tid + st];
        __syncthreads();
    }
    if (tid == 0)
        outaux[0] = 0.01f * buf[0] / ((float)NTOK * (float)NEXP * (float)NEXP);
}

// ---------------------------------------------------------------------------
extern "C" void kernel_launch(void* const* d_in, const int* in_sizes, int n_in,
                              void* d_out, int out_size, void* d_ws, size_t ws_size,
                              hipStream_t stream)
{
    const float* x  = (const float*)d_in[0];   // [2,2048,1024]
    const float* gw = (const float*)d_in[1];   // [1024,8]
    const float* gb = (const float*)d_in[2];   // [8]
    const float* w1 = (const float*)d_in[3];   // [8,1024,2048]
    const float* w2 = (const float*)d_in[4];   // [8,2048,1024]

    float* out    = (float*)d_out;                    // [NTOK*DDIM] then aux scalar
    float* outaux = out + (size_t)NTOK * DDIM;

    // Workspace carve-up (~33.9 MB): contrib planes first (16B aligned).
    float* contrib = (float*)d_ws;                              // 2*NTOK*DDIM f32
    char*  p = (char*)d_ws + (size_t)2 * NTOK * DDIM * sizeof(float);
    int*   idx0   = (int*)p;   p += (size_t)NTOK * 4;
    int*   idx1   = (int*)p;   p += (size_t)NTOK * 4;
    float* s0v    = (float*)p; p += (size_t)NTOK * 4;
    float* s1v    = (float*)p; p += (size_t)NTOK * 4;
    float* auxtok = (float*)p; p += (size_t)NTOK * 4;
    int*   counts = (int*)p;   p += 64;
    int*   list   = (int*)p;   p += (size_t)NEXP * NTOK * 4;
    float* lsc    = (float*)p;

    gate_kernel<<<NTOK, 128, 0, stream>>>(x, gw, gb, idx0, idx1, s0v, s1v, auxtok);
    route_kernel<<<NEXP, 256, 0, stream>>>(idx0, idx1, s0v, s1v, counts, list, lsc);
    expert_kernel<<<NEXP * 256, 128, 0, stream>>>(x, w1, w2, counts, list, lsc, contrib);
    combine_kernel<<<(NTOK * DDIM / 4) / 256, 256, 0, stream>>>(contrib, out);
    aux_kernel<<<1, 256, 0, stream>>>(auxtok, outaux);
}